// HungarianLoss_40518721470850
// MI455X (gfx1250) — compile-verified
//
#include <hip/hip_runtime.h>
#include <math.h>

#define BB   64
#define MM   100
#define NN   100
#define NSQ  100          // Hungarian square size: max(M,K) == M == 100 always
#define PAD  112          // 7 tiles * 16
#define BLK  128
#define FINF 3.402823466e38f

typedef float v2f __attribute__((ext_vector_type(2)));
typedef float v8f __attribute__((ext_vector_type(8)));

__global__ __launch_bounds__(BLK)
void hungarian_loss_kernel(const float* __restrict__ pred,
                           const float* __restrict__ tgt,
                           float* __restrict__ partial)
{
    __shared__ float sq[NSQ * NSQ];                 // 40 KB cost matrix
    __shared__ float px[PAD], py[PAD], pcf[PAD];    // preds (zero padded)
    __shared__ float cx[PAD], cy[PAD], ccf[PAD];    // compacted valid targets
    __shared__ float su[NSQ + 1], sv[NSQ + 1], minv[NSQ + 1];
    __shared__ int   sp[NSQ + 1], way[NSQ + 1], used[NSQ + 1];
    __shared__ float rval[BLK];  __shared__ int ridx[BLK];
    __shared__ float rs0[BLK], rs1[BLK];
    __shared__ int   sK, sj0;

    const int tid = threadIdx.x;
    const int b   = blockIdx.x;
    const float* pb = pred + b * (MM * 3);
    const float* tb = tgt  + b * (NN * 3);

    // ---- phase 1: load preds, zero staging + cost matrix ----
    for (int i = tid; i < PAD; i += BLK) {
        float x = 0.f, y = 0.f, c = 0.f;
        if (i < MM) { x = pb[i * 3 + 0]; y = pb[i * 3 + 1]; c = pb[i * 3 + 2]; }
        px[i] = x; py[i] = y; pcf[i] = c;
        cx[i] = 0.f; cy[i] = 0.f; ccf[i] = 0.f;
    }
    for (int i = tid; i < NSQ * NSQ; i += BLK) sq[i] = 0.f;
    __syncthreads();

    // ---- phase 2: serial compaction of valid targets (conf > 0.5) ----
    if (tid == 0) {
        int k = 0;
        for (int t = 0; t < NN; ++t) {
            float cf = tb[t * 3 + 2];
            if (cf > 0.5f) { cx[k] = tb[t * 3 + 0]; cy[k] = tb[t * 3 + 1]; ccf[k] = cf; ++k; }
        }
        sK = k;
    }
    __syncthreads();
    const int K = sK;

    if (K > 0) {
        // ---- phase 3: cost matrix via V_WMMA_F32_16X16X4_F32 ----
        // dot[m][n] = px[m]*cx[n] + py[m]*cy[n]; cost = sqrt(pp + tt - 2*dot)
        // A (16x4 f32): lanes 0-15 -> K=0,1 = (x,y); lanes 16-31 -> K=2,3 = 0
        // B (4x16 f32): lanes 0-15 -> K=0,1 = (x,y); lanes 16-31 -> K=2,3 = 0
        const int wave = tid >> 5;
        const int lane = tid & 31;
        const int l16  = lane & 15;
        const bool lo  = lane < 16;
        for (int t = wave; t < 49; t += 4) {
            const int ti = t / 7, tj = t % 7;
            v2f a, bf;
            a.x  = lo ? px[ti * 16 + l16] : 0.f;
            a.y  = lo ? py[ti * 16 + l16] : 0.f;
            bf.x = lo ? cx[tj * 16 + l16] : 0.f;
            bf.y = lo ? cy[tj * 16 + l16] : 0.f;
            v8f acc = {0.f, 0.f, 0.f, 0.f, 0.f, 0.f, 0.f, 0.f};
            acc = __builtin_amdgcn_wmma_f32_16x16x4_f32(
                false, a, false, bf, (short)0, acc, false, false);
            // D layout: VGPR r -> M = 16*ti + r (+8 for lanes 16-31), N = 16*tj + lane%16
            const int n = tj * 16 + l16;
            const int mbase = ti * 16 + (lo ? 0 : 8);
            if (n < K) {
                const float tt = cx[n] * cx[n] + cy[n] * cy[n];
#pragma unroll
                for (int r = 0; r < 8; ++r) {
                    const int m = mbase + r;
                    if (m < MM) {
                        const float pp = px[m] * px[m] + py[m] * py[m];
                        const float d2 = pp + tt - 2.f * acc[r];
                        sq[m * NSQ + n] = sqrtf(fmaxf(d2, 0.f));
                    }
                }
            }
        }
        __syncthreads();

        // ---- phase 4: Hungarian (serial over rows, parallel over columns) ----
        if (tid <= NSQ) { su[tid] = 0.f; sv[tid] = 0.f; sp[tid] = 0; way[tid] = 0; }
        __syncthreads();
        for (int i = 1; i <= NSQ; ++i) {
            if (tid == 0) { sp[0] = i; sj0 = 0; }
            if (tid <= NSQ) { minv[tid] = FINF; used[tid] = 0; }
            for (;;) {
                __syncthreads();                        // A: sj0/sp/su/sv consistent
                const int j0 = sj0;
                if (tid == j0) used[tid] = 1;
                __syncthreads();                        // B
                const int   i0 = sp[j0];
                const float uu = su[i0];
                float cv = FINF; int cj = 0x7fffffff;
                if (tid >= 1 && tid <= NSQ && !used[tid]) {
                    float cur = sq[(i0 - 1) * NSQ + (tid - 1)] - uu - sv[tid];
                    if (cur < minv[tid]) { minv[tid] = cur; way[tid] = j0; }
                    cv = minv[tid]; cj = tid;
                }
                rval[tid] = cv; ridx[tid] = cj;
                __syncthreads();                        // C
                for (int s = BLK / 2; s > 0; s >>= 1) { // argmin, ties -> smallest j
                    if (tid < s) {
                        float v2 = rval[tid + s]; int i2 = ridx[tid + s];
                        if (v2 < rval[tid] || (v2 == rval[tid] && i2 < ridx[tid])) {
                            rval[tid] = v2; ridx[tid] = i2;
                        }
                    }
                    __syncthreads();
                }
                const int   j1    = ridx[0];
                const float delta = rval[0];
                if (tid <= NSQ) {
                    if (used[tid]) { su[sp[tid]] += delta; sv[tid] -= delta; }
                    else           { minv[tid] -= delta; }
                }
                if (tid == 0) sj0 = j1;
                __syncthreads();                        // D
                if (sp[j1] == 0) break;                 // uniform condition
            }
            if (tid == 0) {                             // augment along way[]
                int j0l = sj0;
                while (j0l) { int jn = way[j0l]; sp[j0l] = sp[jn]; j0l = jn; }
            }
        }
        __syncthreads();

        // ---- phase 5: loss over matched pairs (col j -> row sp[j]-1) ----
        float cs = 0.f, bs = 0.f;
        if (tid >= 1 && tid <= K) {
            const int r = sp[tid] - 1;                  // pred row
            const int c = tid - 1;                      // compact target index
            const float dx = px[r] - cx[c];
            const float dy = py[r] - cy[c];
            cs = dx * dx + dy * dy;
            float pcv = fminf(fmaxf(pcf[r], 1e-12f), 1.0f);
            const float tcv = ccf[c];
            bs = -(tcv * logf(pcv) + (1.f - tcv) * log1pf(-pcv));
        }
        rs0[tid] = cs; rs1[tid] = bs;
        __syncthreads();
        for (int s = BLK / 2; s > 0; s >>= 1) {
            if (tid < s) { rs0[tid] += rs0[tid + s]; rs1[tid] += rs1[tid + s]; }
            __syncthreads();
        }
        if (tid == 0) partial[b] = rs0[0] / (2.f * (float)K) + rs1[0] / (float)K;
    } else {
        if (tid == 0) partial[b] = 0.f;
    }
}

// Deterministic fixed-order final reduction (no float atomics).
__global__ void finalize_kernel(const float* __restrict__ partial,
                                float* __restrict__ out)
{
    if (threadIdx.x == 0 && blockIdx.x == 0) {
        float s = 0.f;
        for (int i = 0; i < BB; ++i) s += partial[i];
        out[0] = s / (float)BB;
    }
}

extern "C" void kernel_launch(void* const* d_in, const int* in_sizes, int n_in,
                              void* d_out, int out_size, void* d_ws, size_t ws_size,
                              hipStream_t stream) {
    (void)in_sizes; (void)n_in; (void)out_size; (void)ws_size;
    const float* pred = (const float*)d_in[0];   // [64,100,3] f32
    const float* tgt  = (const float*)d_in[1];   // [64,100,3] f32
    float* out        = (float*)d_out;           // scalar f32
    float* partial    = (float*)d_ws;            // 64 f32 scratch

    hungarian_loss_kernel<<<BB, BLK, 0, stream>>>(pred, tgt, partial);
    finalize_kernel<<<1, 32, 0, stream>>>(partial, out);
}